// ES_2817498546595
// MI455X (gfx1250) — compile-verified
//
#include <hip/hip_runtime.h>

// Problem constants (match reference)
#define BB 16384
#define TT 512
#define MM 128               // seasonality period == block size for the scan
#define HH 32
#define NN (BB + TT - 1)     // 16895 scan steps
#define NBLK ((NN + MM - 1) / MM)  // 132 blocks

typedef float v2f __attribute__((ext_vector_type(2)));
typedef float v8f __attribute__((ext_vector_type(8)));

// Fast reciprocal: v_rcp_f32 + one Newton-Raphson step (~1 ulp).
__device__ __forceinline__ float fastrcp(float x) {
    float r = __builtin_amdgcn_rcpf(x);
    float e = __builtin_fmaf(-x, r, 1.0f);
    return __builtin_fmaf(r, e, r);
}

// ---------------------------------------------------------------------------
// Phase 1: the coupled level/seasonality scan, one wave32, WMMA-accelerated.
//
// Per 128-step block (valid because the seasonality lag is exactly MM=128):
//   c_t = a*y_t/S[t]           (parallel: S[t] written >=128 steps ago)
//   l   = T128 * c + carry     (lower-triangular scan, done as D = T16 x C
//                               with the 8 sixteen-step segments as columns
//                               of B -> 4x V_WMMA_F32_16X16X4_F32, then an
//                               8-step cross-column carry fix via shuffles)
//   S[t+128] = g*y_t/l_t + (1-g)*S[t]
//   xnorm[t] = y_t/(S[t]*l_t)
// ---------------------------------------------------------------------------
__global__ __launch_bounds__(32) void es_scan_kernel(
    const float* __restrict__ x,
    const float* __restrict__ alpha,
    const float* __restrict__ gamma,
    const float* __restrict__ init_seas,
    const float* __restrict__ level0,
    float* __restrict__ xnorm,    // NN floats
    float* __restrict__ levels,   // NN floats
    float* __restrict__ Sout)     // NN floats: Sout[t] = S[MM + t]
{
    __shared__ float S_lds[256];  // circular window of seasonality (lag 128)
    __shared__ float c_lds[256];  // per-block c values; upper half stays ZERO
    __shared__ float l_lds[256];  // per-block level values; upper half unused
    __shared__ float bpow[17];    // b^0 .. b^16

    const int lane = threadIdx.x;        // wave32
    const float a  = alpha[0];
    const float g  = gamma[0];
    const float b  = 1.0f - a;
    const float gc = 1.0f - g;

    if (lane < 17) {
        float p = 1.0f;
        for (int i = 0; i < lane; ++i) p *= b;
        bpow[lane] = p;
    }
    for (int i = lane; i < MM; i += 32) {
        S_lds[i]      = init_seas[i];
        c_lds[MM + i] = 0.0f;            // zero pad: B columns 8..15 read this
    }
    __syncthreads();
    const float b16 = bpow[16];

    // Constant A fragments: T16[m][k] = (k<=m) ? b^(m-k) : 0, K chunked by 4.
    // A 16x4 f32 layout: lanes 0-15 hold K=0(v0),K=1(v1); lanes 16-31 K=2,K=3.
    const int m     = lane & 15;
    const int khalf = (lane >> 4) * 2;
    v2f Afrag[4];
#pragma unroll
    for (int c = 0; c < 4; ++c) {
        int k0 = 4 * c + khalf;
        Afrag[c].x = (k0     <= m) ? bpow[m - k0]     : 0.0f;
        Afrag[c].y = (k0 + 1 <= m) ? bpow[m - k0 - 1] : 0.0f;
    }

    float l_carry = level0[0];

    for (int blk = 0; blk < NBLK; ++blk) {
        const int base = blk * MM;

        // ---- parallel precompute of y, 1/s, c for this block ----
        float yv[4], sv[4], isv[4];
#pragma unroll
        for (int i = 0; i < 4; ++i) {
            int u = lane + 32 * i;            // step offset within block
            int t = base + u;
            float y;
            if (t < NN) y = (t < TT) ? x[t] : x[(t - TT + 1) * TT + (TT - 1)];
            else        y = 1.0f;             // pad (never stored)
            float s  = S_lds[t & 255];
            float is = fastrcp(s);
            yv[i] = y; sv[i] = s; isv[i] = is;
            c_lds[u] = a * y * is;
        }
        __syncthreads();

        // ---- segment-local scans via WMMA: D = T16 x C (8 used columns,
        //      columns 8..15 read the permanently-zero pad region) ----
        // B 4x16 f32 layout: vgpr v, lanes 0-15 -> K=v; lanes 16-31 -> K=v+2.
        const int j = lane & 15;              // column (= segment index)
        v8f acc = {0.f, 0.f, 0.f, 0.f, 0.f, 0.f, 0.f, 0.f};
#pragma unroll
        for (int c = 0; c < 4; ++c) {
            int k0 = 4 * c + khalf;
            int ci = ((16 * j + k0) & 255);   // j>=8 lands in the zero pad
            v2f Bf;
            Bf.x = c_lds[ci];
            Bf.y = c_lds[ci + 1];
            acc = __builtin_amdgcn_wmma_f32_16x16x4_f32(
                false, Afrag[c], false, Bf, (short)0, acc, false, false);
        }

        // ---- cross-segment carry fix (8 serial fmas, uniform on wave) ----
        // D[15][j] lives in vgpr7, lane 16+j.
        float mycar = l_carry;                // carry INTO column j
        float cur   = l_carry;
#pragma unroll
        for (int jj = 0; jj < 8; ++jj) {
            float zj = __shfl(acc[7], 16 + jj);
            if (j == jj) mycar = cur;
            cur = __builtin_fmaf(b16, cur, zj);   // Lc_{jj+1}
        }
        l_carry = cur;                        // l at end of block

        // ---- apply carry; scatter l into LDS in step order (unconditional;
        //      j>=8 writes land in the unused upper half) ----
        // D layout: vgpr v, lanes 0-15 -> row v; lanes 16-31 -> row v+8.
        {
            const int mh = 8 * (lane >> 4);
#pragma unroll
            for (int v = 0; v < 8; ++v) {
                int mrow = v + mh;
                float lv = __builtin_fmaf(bpow[mrow + 1], mycar, acc[v]);
                l_lds[16 * j + mrow] = lv;
            }
        }
        __syncthreads();

        // ---- finalize: seasonality update + outputs ----
#pragma unroll
        for (int i = 0; i < 4; ++i) {
            int u = lane + 32 * i;
            int t = base + u;
            float lv = l_lds[u];
            float il = fastrcp(lv);
            float y = yv[i], s = sv[i], is = isv[i];
            float snew = __builtin_fmaf(g * y, il, gc * s);
            S_lds[(t + MM) & 255] = snew;     // visible to next block
            if (t < NN) {
                Sout[t]   = snew;
                xnorm[t]  = (y * is) * il;
                levels[t] = lv;
            }
        }
        __syncthreads();
    }
}

// ---------------------------------------------------------------------------
// Phase 2: x_out[b,t] = xnorm[b+t]  (write-bandwidth bound, 33.5 MB; the
// 68 KB xnorm source stays resident in L2)
// ---------------------------------------------------------------------------
__global__ void es_gather_kernel(const float* __restrict__ xnorm,
                                 float* __restrict__ x_out)
{
    long idx = ((long)blockIdx.x * blockDim.x + threadIdx.x) * 4;
    if (idx >= (long)BB * TT) return;
    int bidx = (int)(idx >> 9);          // TT = 512
    int t    = (int)(idx & (TT - 1));    // multiple of 4
    int src  = bidx + t;
    float4 v = make_float4(xnorm[src], xnorm[src + 1],
                           xnorm[src + 2], xnorm[src + 3]);
    *(float4*)(x_out + idx) = v;
}

// ---------------------------------------------------------------------------
// Phase 3: denorm[b,h] = (levels[T-1+b], S[T+b+h]);  S[j] = Sout[j-MM]
// ---------------------------------------------------------------------------
__global__ void es_denorm_kernel(const float* __restrict__ levels,
                                 const float* __restrict__ Sout,
                                 float* __restrict__ out)
{
    int i = blockIdx.x * blockDim.x + threadIdx.x;   // i = b*HH + h
    if (i >= BB * HH) return;
    int bidx = i >> 5;                               // HH = 32
    int h    = i & 31;
    float2 v;
    v.x = levels[TT - 1 + bidx];
    v.y = Sout[(TT - MM) + bidx + h];
    *(float2*)(out + 2L * i) = v;
}

extern "C" void kernel_launch(void* const* d_in, const int* in_sizes, int n_in,
                              void* d_out, int out_size, void* d_ws, size_t ws_size,
                              hipStream_t stream)
{
    const float* x         = (const float*)d_in[0];
    const float* alpha     = (const float*)d_in[1];
    const float* gamma     = (const float*)d_in[2];
    const float* init_seas = (const float*)d_in[3];
    const float* level     = (const float*)d_in[4];
    float* out = (float*)d_out;
    float* ws  = (float*)d_ws;

    float* xnorm  = ws;            // NN floats
    float* levels = ws + NN;       // NN floats
    float* Sout   = ws + 2 * NN;   // NN floats   (~200 KB total)

    es_scan_kernel<<<1, 32, 0, stream>>>(x, alpha, gamma, init_seas, level,
                                         xnorm, levels, Sout);

    long nvec = (long)BB * TT / 4;
    es_gather_kernel<<<(int)((nvec + 255) / 256), 256, 0, stream>>>(xnorm, out);

    es_denorm_kernel<<<(BB * HH + 255) / 256, 256, 0, stream>>>(
        levels, Sout, out + (long)BB * TT);
}